// snn_model_26895085207938
// MI455X (gfx1250) — compile-verified
//
#include <hip/hip_runtime.h>

typedef __attribute__((ext_vector_type(2))) float v2f;
typedef __attribute__((ext_vector_type(8))) float v8f;

#define TSTEPS 20
#define DECAY  0.9375f

// workspace layout (float offsets)
#define OFF_SIN  0u          // 20*3*64*64   = 245760  input spikes  [t][c][y][x]
#define OFF_S1   245760u     // 20*32*60*60  = 2304000 conv1 spikes consumed by conv2 at step t
#define OFF_S2   2549760u    // 20*200704    = 4014080 conv2 spikes consumed by linear1 at step t
#define OFF_D2   6563840u    // 20*64*56*56  = 4014080 conv2 drives  [t][ch][sp]
#define OFF_L1D  10577920u   // 512*32                 linear1 drives [m][t(pad32)]

// ---------------- phase 0: avgpool + input integrate-and-fire (no decay) ----
__global__ void k_input(const float* __restrict__ image, float* __restrict__ ws) {
  int p = blockIdx.x * blockDim.x + threadIdx.x;
  if (p >= 3 * 64 * 64) return;
  int c = p >> 12, r = p & 4095, y = r >> 6, x = r & 63;
  const float* im = image + (size_t)(c * 128 + 2 * y) * 128 + 2 * x;
  float v = 0.25f * (im[0] + im[1] + im[128] + im[129]);
  float* sin_ = ws + OFF_SIN;
  float mp = 0.f;
  for (int t = 0; t < TSTEPS; ++t) {
    mp += v;
    float s = (mp >= 1.f) ? 1.f : 0.f;
    mp -= s;
    sin_[t * 12288 + p] = s;
  }
}

// ---------------- phase 1: conv1 drive + LIF fused, per output element ------
// writes S1[t] = conv1 spikes from step t-1 (S1[0] = 0); only t=0..18 needed.
__global__ void k_conv1(const float* __restrict__ w_c1, float* __restrict__ ws) {
  int e = blockIdx.x * blockDim.x + threadIdx.x;
  if (e >= 32 * 60 * 60) return;
  int ch = e / 3600, r = e % 3600, oy = r / 60, ox = r % 60;
  const float* sin_ = ws + OFF_SIN;
  float* S1 = ws + OFF_S1;
  const float* w = w_c1 + ch * 75;
  S1[e] = 0.f;  // S1[0]
  float mp = 0.f;
  for (int t = 0; t < TSTEPS - 1; ++t) {
    const float* st = sin_ + t * 12288;
    float d = 0.f;
#pragma unroll
    for (int c = 0; c < 3; ++c)
#pragma unroll
      for (int dy = 0; dy < 5; ++dy)
#pragma unroll
        for (int dx = 0; dx < 5; ++dx)
          d += st[c * 4096 + (oy + dy) * 64 + (ox + dx)] * w[c * 25 + dy * 5 + dx];
    mp += d;
    float s = (mp >= 1.f) ? 1.f : 0.f;
    mp = (s > 0.f) ? (mp - 1.f) : mp * DECAY;
    S1[(t + 1) * 115200 + e] = s;
  }
}

// ---------------- phase 2: conv2 as implicit GEMM via V_WMMA_F32_16X16X4_F32
// M = 64 out channels (4 tiles, one per wave), N = 3136 spatial (196 tiles),
// K = 32*25 = 800 (200 wmma steps).  Batched over t = 1..19 (t=0 drive is 0).
__global__ void __launch_bounds__(128) k_conv2_wmma(const float* __restrict__ w_c2,
                                                    float* __restrict__ ws) {
  const float* S1 = ws + OFF_S1;
  float* D2 = ws + OFF_D2;
  int t = blockIdx.y + 1;              // 1..19
  int nt = blockIdx.x;                 // spatial tile
  int lane = threadIdx.x & 31;
  int wv = threadIdx.x >> 5;           // channel tile 0..3
  int kh = (lane >> 4) << 1;           // K sub-offset per A/B layout (0 or 2)
  int m = (wv << 4) + (lane & 15);     // out channel (A row)
  int nsp = (nt << 4) + (lane & 15);   // spatial position (B column)
  int oy = nsp / 56, ox = nsp % 56;
  const float* wrow = w_c2 + (size_t)m * 800;
  const float* s1t = S1 + (size_t)t * 115200;
  v8f acc = {};
  for (int k = 0; k < 800; k += 4) {
    int ka = k + kh, kb = ka + 1;
    v2f a; a.x = wrow[ka]; a.y = wrow[kb];
    int ca = ka / 25, ra = ka % 25, dya = ra / 5, dxa = ra % 5;
    int cb = kb / 25, rb = kb % 25, dyb = rb / 5, dxb = rb % 5;
    v2f b;
    b.x = s1t[ca * 3600 + (oy + dya) * 60 + (ox + dxa)];
    b.y = s1t[cb * 3600 + (oy + dyb) * 60 + (ox + dxb)];
    acc = __builtin_amdgcn_wmma_f32_16x16x4_f32(false, a, false, b, (short)0, acc,
                                                false, false);
  }
  int mi = (lane >> 4) << 3;  // +8 row offset for upper half-wave (C/D layout)
  float* dout = D2 + (size_t)t * 200704;
#pragma unroll
  for (int i = 0; i < 8; ++i) {
    int mm = (wv << 4) + mi + i;
    dout[(size_t)mm * 3136 + nsp] = acc[i];
  }
}

// ---------------- phase 3: conv2 LIF, produce S2 (delayed spikes) -----------
__global__ void k_conv2_lif(float* __restrict__ ws) {
  int e = blockIdx.x * blockDim.x + threadIdx.x;
  if (e >= 200704) return;
  const float* D2 = ws + OFF_D2;
  float* S2 = ws + OFF_S2;
  S2[e] = 0.f;  // S2[0]
  float mp = 0.f;
  for (int t = 0; t < TSTEPS - 1; ++t) {
    float d = (t == 0) ? 0.f : D2[(size_t)t * 200704 + e];
    mp += d;
    float s = (mp >= 1.f) ? 1.f : 0.f;
    mp = (s > 0.f) ? (mp - 1.f) : mp * DECAY;
    S2[(size_t)(t + 1) * 200704 + e] = s;
  }
}

__global__ void k_zero_l1d(float* __restrict__ ws) {
  int i = blockIdx.x * blockDim.x + threadIdx.x;
  if (i < 512 * 32) ws[OFF_L1D + i] = 0.f;
}

// ---------------- phase 4: linear1 as one GEMM over all timesteps -----------
// D(512 x 32pad) = W(512 x 200704) * S2^T(200704 x 32pad), split-K (392 x 512)
// with f32 atomic accumulation. Reads w_l1 exactly once (~411 MB -> ~18us HBM).
__global__ void __launch_bounds__(64) k_lin1_wmma(const float* __restrict__ w_l1,
                                                  float* __restrict__ ws) {
  const float* S2 = ws + OFF_S2;
  float* L1D = ws + OFF_L1D;
  int mt = blockIdx.x;                 // M tile 0..31
  int kc = blockIdx.y;                 // K chunk 0..391
  int lane = threadIdx.x & 31;
  int wv = threadIdx.x >> 5;           // N tile 0..1
  int kh = (lane >> 4) << 1;
  int m = (mt << 4) + (lane & 15);     // W row
  int n = (wv << 4) + (lane & 15);     // padded timestep column
  int nc = (n < TSTEPS) ? n : (TSTEPS - 1);
  float nmask = (n < TSTEPS) ? 1.f : 0.f;
  const float* wrow = w_l1 + (size_t)m * 200704 + (size_t)kc * 512;
  const float* scol = S2 + (size_t)nc * 200704 + (size_t)kc * 512;
  v8f acc = {};
  for (int k = 0; k < 512; k += 4) {
    v2f a; a.x = wrow[k + kh]; a.y = wrow[k + kh + 1];
    v2f b; b.x = scol[k + kh] * nmask; b.y = scol[k + kh + 1] * nmask;
    acc = __builtin_amdgcn_wmma_f32_16x16x4_f32(false, a, false, b, (short)0, acc,
                                                false, false);
  }
  int mi = (lane >> 4) << 3;
#pragma unroll
  for (int i = 0; i < 8; ++i) {
    int mm = (mt << 4) + mi + i;
    int nn = (wv << 4) + (lane & 15);
    atomicAdd(&L1D[mm * 32 + nn], acc[i]);
  }
}

// ---------------- phase 5: linear1 LIF + linear2 head, one block ------------
__global__ void __launch_bounds__(512) k_head(const float* __restrict__ w_l2,
                                              float* __restrict__ out,
                                              const float* __restrict__ ws) {
  __shared__ float sp[512];   // ns_l1(t-1)
  const float* L1D = ws + OFF_L1D;
  int m = threadIdx.x;
  float mp1 = 0.f, mp2 = 0.f;
  sp[m] = 0.f;
  if (m < 10) out[m] = 0.f;   // output row 0 stays zero
  __syncthreads();
  for (int t = 0; t < TSTEPS; ++t) {
    // linear2 consumes sp = ns_l1(t-1)
    if (m < 10) {
      float d = 0.f;
      for (int j = 0; j < 512; ++j) d += w_l2[m * 512 + j] * sp[j];
      mp2 += d;
      float s2 = (mp2 >= 1.f) ? 1.f : 0.f;
      mp2 = (s2 > 0.f) ? (mp2 - 1.f) : mp2 * DECAY;
      out[(t + 1) * 10 + m] = s2;
    }
    // linear1 LIF for this step
    float d1 = L1D[m * 32 + t];
    mp1 += d1;
    float s1 = (mp1 >= 1.f) ? 1.f : 0.f;
    mp1 = (s1 > 0.f) ? (mp1 - 1.f) : mp1 * DECAY;
    __syncthreads();
    sp[m] = s1;
    __syncthreads();
  }
}

extern "C" void kernel_launch(void* const* d_in, const int* in_sizes, int n_in,
                              void* d_out, int out_size, void* d_ws, size_t ws_size,
                              hipStream_t stream) {
  (void)in_sizes; (void)n_in; (void)out_size; (void)ws_size;
  const float* image = (const float*)d_in[0];
  const float* w_c1  = (const float*)d_in[1];
  const float* w_c2  = (const float*)d_in[2];
  const float* w_l1  = (const float*)d_in[3];
  const float* w_l2  = (const float*)d_in[4];
  float* out = (float*)d_out;
  float* ws  = (float*)d_ws;

  k_input    <<<dim3(48),       256, 0, stream>>>(image, ws);
  k_conv1    <<<dim3(450),      256, 0, stream>>>(w_c1, ws);
  k_conv2_wmma<<<dim3(196, 19), 128, 0, stream>>>(w_c2, ws);
  k_conv2_lif<<<dim3(784),      256, 0, stream>>>(ws);
  k_zero_l1d <<<dim3(64),       256, 0, stream>>>(ws);
  k_lin1_wmma<<<dim3(32, 392),   64, 0, stream>>>(w_l1, ws);
  k_head     <<<dim3(1),        512, 0, stream>>>(w_l2, out, ws);
}